// sym_model_65171833749980
// MI455X (gfx1250) — compile-verified
//
#include <hip/hip_runtime.h>

// Lattice-gauge symmetric CNN for MI455X (gfx1250, wave32, WMMA).
//
// Symmetry shortcut: periodic conv + ReLU is translation-equivariant and the
// sum-pool is translation-invariant, so all 100 cyclic shifts give identical
// pooled outputs -> mean over shifts == one un-shifted forward pass (100x
// fewer FLOPs than the literal reference, equal to fp32 rounding).  The
// remaining ~1 GFLOP / ~350 KB problem sits far under both the 23.3 TB/s HBM
// and the WMMA roofline, so we keep full fp32 precision via
// V_WMMA_F32_16X16X4_F32.
//
// Structure: one workgroup (8 wave32) per sample; each conv layer is an
// implicit GEMM (M=128 padded sites, N=32 channels, K=Cin*9) with per-wave
// 16x32 D blocks.  Wrapped neighbor sites are precomputed once per lane
// (wsite[10], register-resident under full unroll); A and B fragments are
// single ds_load_b64 each via a pair-interleaved LDS weight layout.  Padding
// uses a permanently-zero row for A-reads and a min-clamped garbage row for
// D-stores, so the WMMA region and the stores have NO divergent branches
// (EXEC stays all-ones as the ISA requires).

typedef __attribute__((ext_vector_type(2))) float v2f;
typedef __attribute__((ext_vector_type(8))) float v8f;

#define LAT   10
#define NSITE 100                  // 10*10 lattice sites
#define FCH   32                   // hidden channels
#define KMAX  288                  // 32 channels * 9 taps
#define GROW  100                  // garbage row: clamp target for pad stores
#define ZROW  101                  // always-zero row: A-read padding
#define AROWS 102

static __device__ __forceinline__ v8f wmma_f32(v2f a, v2f b, v8f c) {
    return __builtin_amdgcn_wmma_f32_16x16x4_f32(false, a, false, b,
                                                 (short)0, c, false, false);
}

// Bias + ReLU + store one 16x32 D block (two 16x16 tiles).  Out-of-range rows
// are clamped to the garbage row GROW (never read) -> unconditional stores.
static __device__ __forceinline__ void store_tiles(float* __restrict__ actOut,
                                                   const float* __restrict__ bias,
                                                   const v8f& acc0, const v8f& acc1,
                                                   int wave, int lane) {
    const int n  = lane & 15;
    const int hi = lane >> 4;
    const float bn0 = bias[n], bn1 = bias[16 + n];
#pragma unroll
    for (int r = 0; r < 8; ++r) {
        int srow = wave * 16 + r + hi * 8;              // D: vgpr r -> M=r(+8)
        srow = srow < GROW ? srow : GROW;               // clamp, don't branch
        float v0 = acc0[r] + bn0;
        float v1 = acc1[r] + bn1;
        actOut[srow * FCH + n]      = v0 > 0.f ? v0 : 0.f;
        actOut[srow * FCH + 16 + n] = v1 > 0.f ? v1 : 0.f;
    }
}

// 32->32 periodic 3x3 conv as implicit GEMM, M=128(pad) N=32 K=288.
// k = tap*32 + c ; per wave: one M tile, both N tiles.
static __device__ __forceinline__ void conv32(const float* __restrict__ actIn,
                                              float* __restrict__ actOut,
                                              const float* __restrict__ wpair,
                                              const float* __restrict__ bias,
                                              int wave, int lane,
                                              const int* wsite) {
    const int n  = lane & 15;
    const int hi = lane >> 4;
    const float* aLane = actIn + hi * 2;           // A: lanes>=16 hold K{2,3}
    const float* bLane = wpair + n * 2 + hi * 64;  // pair idx = tap*16+t*2+hi

    v8f acc0 = {0.f, 0.f, 0.f, 0.f, 0.f, 0.f, 0.f, 0.f};
    v8f acc1 = {0.f, 0.f, 0.f, 0.f, 0.f, 0.f, 0.f, 0.f};

#pragma unroll
    for (int tap = 0; tap < 9; ++tap) {
        const float* aBase = aLane + wsite[tap] * FCH;
#pragma unroll
        for (int t = 0; t < 8; ++t) {              // k0 = tap*32 + t*4 + hi*2
            v2f a  = *(const v2f*)(aBase + t * 4);                     // b64
            v2f b0 = *(const v2f*)(bLane + (tap * 16 + t * 2) * 64);   // b64
            v2f b1 = *(const v2f*)(bLane + (tap * 16 + t * 2) * 64 + 32);
            acc0 = wmma_f32(a, b0, acc0);
            acc1 = wmma_f32(a, b1, acc1);
        }
    }
    store_tiles(actOut, bias, acc0, acc1, wave, lane);
}

// 2->32 first layer, K = 18 padded to 20 (5 ksteps); k = tap*2 + c.
static __device__ __forceinline__ void conv2ch(const float* __restrict__ actIn,
                                               float* __restrict__ actOut,
                                               const float* __restrict__ wpair,
                                               const float* __restrict__ bias,
                                               int wave, int lane,
                                               const int* wsite) {
    const int n  = lane & 15;
    const int hi = lane >> 4;
    const float* bLane = wpair + n * 2 + hi * 64;  // pair idx = ks*2 + hi

    v8f acc0 = {0.f, 0.f, 0.f, 0.f, 0.f, 0.f, 0.f, 0.f};
    v8f acc1 = {0.f, 0.f, 0.f, 0.f, 0.f, 0.f, 0.f, 0.f};

#pragma unroll
    for (int ks = 0; ks < 5; ++ks) {               // tap = 2*ks + hi (<=9)
        const int ws = hi ? wsite[2 * ks + 1] : wsite[2 * ks];
        v2f a  = *(const v2f*)(actIn + ws * 2);    // both channels of tap
        v2f b0 = *(const v2f*)(bLane + ks * 128);
        v2f b1 = *(const v2f*)(bLane + ks * 128 + 32);
        acc0 = wmma_f32(a, b0, acc0);
        acc1 = wmma_f32(a, b1, acc1);
    }
    store_tiles(actOut, bias, acc0, acc1, wave, lane);
}

__launch_bounds__(256, 1)
__global__ void lgcnn_fwd(const float* __restrict__ x,
                          const float* __restrict__ W1, const float* __restrict__ b1,
                          const float* __restrict__ W2, const float* __restrict__ b2,
                          const float* __restrict__ W3, const float* __restrict__ b3,
                          const float* __restrict__ Wout, const float* __restrict__ bout,
                          float* __restrict__ out) {
    __shared__ float wpair[KMAX * FCH];     // 36864 B, pair-interleaved weights
    __shared__ float actA[AROWS * FCH];     // 13056 B ping (+garbage+zero rows)
    __shared__ float actB[AROWS * FCH];     // 13056 B pong (+garbage+zero rows)
    __shared__ float bias[FCH];
    __shared__ float red[256];

    const int tid  = threadIdx.x;
    const int lane = tid & 31;
    const int wave = tid >> 5;
    const int b    = blockIdx.x;

    // ---- per-lane wrapped neighbor sites (register array, computed once).
    // Padded lanes (site>=100) and the layer-1 pad tap alias the zero row.
    int wsite[10];
    {
        const int site = wave * 16 + (lane & 15);
        const int pad  = site >= NSITE;
        const int si = site / LAT, sj = site - si * LAT;
#pragma unroll
        for (int di = 0; di < 3; ++di)
#pragma unroll
            for (int dj = 0; dj < 3; ++dj) {
                int ii = si + di; if (ii >= LAT) ii -= LAT;
                int jj = sj + dj; if (jj >= LAT) jj -= LAT;
                wsite[di * 3 + dj] = pad ? ZROW : (ii * LAT + jj);
            }
        wsite[9] = ZROW;                    // layer-1 K-pad tap -> zero row
    }

    // ---- zero pad rows + stage input sample x:(B,2,10,10) -> actA[site*2+c]
    if (tid < FCH) { actA[ZROW * FCH + tid] = 0.f; actB[ZROW * FCH + tid] = 0.f; }
    if (tid < 4)   { actA[NSITE * 2 + tid] = 0.f; }   // stride-2 zero pad @202
    for (int i = tid; i < NSITE * 2; i += 256) {
        int site = i >> 1, c = i & 1;
        actA[site * 2 + c] = x[b * (2 * NSITE) + c * NSITE + site];
    }
    // ---- W1 (32,2,3,3) -> wpair[(k>>1)*64 + n*2 + (k&1)], k=tap*2+c, pad->20
    for (int i = tid; i < 20 * FCH; i += 256) {
        int p = i & 1, n = (i >> 1) & 31, k = (i >> 6) * 2 + p;
        float v = 0.f;
        if (k < 18) { int tap = k >> 1, c = k & 1; v = W1[n * 18 + c * 9 + tap]; }
        wpair[i] = v;
    }
    if (tid < FCH) bias[tid] = b1[tid];
    __syncthreads();

    conv2ch(actA, actB, wpair, bias, wave, lane, wsite);        // 2 -> 32
    __syncthreads();

    // ---- W2 (32,32,3,3) -> pair-interleaved, k = tap*32 + c
    for (int i = tid; i < KMAX * FCH; i += 256) {
        int p = i & 1, n = (i >> 1) & 31, k = (i >> 6) * 2 + p;
        int tap = k >> 5, c = k & 31;
        wpair[i] = W2[n * KMAX + c * 9 + tap];
    }
    if (tid < FCH) bias[tid] = b2[tid];
    __syncthreads();

    conv32(actB, actA, wpair, bias, wave, lane, wsite);         // 32 -> 32
    __syncthreads();

    for (int i = tid; i < KMAX * FCH; i += 256) {
        int p = i & 1, n = (i >> 1) & 31, k = (i >> 6) * 2 + p;
        int tap = k >> 5, c = k & 31;
        wpair[i] = W3[n * KMAX + c * 9 + tap];
    }
    if (tid < FCH) bias[tid] = b3[tid];
    __syncthreads();

    conv32(actA, actB, wpair, bias, wave, lane, wsite);         // 32 -> 32
    __syncthreads();

    // ---- sum-pool over sites fused with Wout projection (deterministic tree)
    float local = 0.f;
    for (int i = tid; i < NSITE * FCH; i += 256)
        local += actB[i] * Wout[i & 31];
    red[tid] = local;
    __syncthreads();
#pragma unroll
    for (int s = 128; s > 0; s >>= 1) {
        if (tid < s) red[tid] += red[tid + s];
        __syncthreads();
    }
    if (tid == 0) out[b] = red[0] + bout[0];
}

extern "C" void kernel_launch(void* const* d_in, const int* in_sizes, int n_in,
                              void* d_out, int out_size, void* d_ws, size_t ws_size,
                              hipStream_t stream) {
    const float* x    = (const float*)d_in[0];
    const float* W1   = (const float*)d_in[1];
    const float* b1   = (const float*)d_in[2];
    const float* W2   = (const float*)d_in[3];
    const float* b2   = (const float*)d_in[4];
    const float* W3   = (const float*)d_in[5];
    const float* b3   = (const float*)d_in[6];
    const float* Wout = (const float*)d_in[7];
    const float* bout = (const float*)d_in[8];
    float* out = (float*)d_out;

    const int batch = out_size;  // OUT == 1, one scalar per sample
    lgcnn_fwd<<<batch, 256, 0, stream>>>(x, W1, b1, W2, b2, W3, b3, Wout, bout, out);
}